// GraphSAGE_66915590472496
// MI455X (gfx1250) — compile-verified
//
#include <hip/hip_runtime.h>
#include <hip/hip_bf16.h>

// ---------------------------------------------------------------------------
// GraphSAGE 2-layer, fp32, D=128, on gfx1250 (MI455X).
//   layer: out = mean_agg @ W_l^T + b_l + x_dst @ W_r^T ; L2-normalize ; (relu)
// GEMM part uses V_WMMA_F32_16X16X4_F32 (full fp32 precision).
// Scatter part uses coalesced row gathers + global_atomic_add_f32.
// ---------------------------------------------------------------------------

#define D 128
#define N_MID 50000
#define N_DST 10000

typedef __attribute__((ext_vector_type(2))) float v2f;
typedef __attribute__((ext_vector_type(8))) float v8f;

// ---------------------------------------------------------------------------
// Edge scatter: one wave (32 lanes) per edge; each lane moves 4 floats.
//   agg[col[e]] += src[row[e]] ; deg[col[e]] += 1
// ---------------------------------------------------------------------------
__global__ __launch_bounds__(256) void sage_scatter(
    const float* __restrict__ src, const int* __restrict__ row,
    const int* __restrict__ col, float* __restrict__ agg,
    float* __restrict__ deg, int E) {
  int e = blockIdx.x * 8 + (threadIdx.x >> 5);
  int lane = threadIdx.x & 31;
  if (e >= E) return;
  int r = row[e];
  int c = col[e];
  const float4 v = ((const float4*)(src + (size_t)r * D))[lane];
  float* dst = agg + (size_t)c * D + lane * 4;
  unsafeAtomicAdd(dst + 0, v.x);
  unsafeAtomicAdd(dst + 1, v.y);
  unsafeAtomicAdd(dst + 2, v.z);
  unsafeAtomicAdd(dst + 3, v.w);
  if (lane == 0) unsafeAtomicAdd(deg + c, 1.0f);
}

// ---------------------------------------------------------------------------
// Fused SAGE layer GEMM:
//   out[m,:] = normalize( mean[m,:] @ Wl^T + bl + xdst[m,:] @ Wr^T )  (opt relu)
// Block = 256 threads = 8 waves; block handles a 16-row M tile; wave w handles
// the 16-column N tile [w*16, w*16+16).  K = 128 -> 32 WMMA f32_16x16x4 steps
// per weight matrix.
// M must be a multiple of 16 (50000 and 10000 both are).
// ---------------------------------------------------------------------------
template <int RELU>
__global__ __launch_bounds__(256) void sage_gemm(
    const float* __restrict__ agg, const float* __restrict__ deg,
    const float* __restrict__ xdst, const float* __restrict__ Wl,
    const float* __restrict__ bl, const float* __restrict__ Wr,
    float* __restrict__ out, int M) {
  __shared__ float sMean[16][D + 4];  // +4 pad: kills LDS bank conflicts
  __shared__ float sDst[16][D + 4];
  __shared__ float sOut[16][D];
  __shared__ float sInv[16];
  __shared__ float sNorm[16];

  const int tid = threadIdx.x;
  const int lane = tid & 31;
  const int w = tid >> 5;             // wave id = N tile
  const int mtile = blockIdx.x;       // 16-row tile
  const int m0 = mtile * 16;

  if (tid < 16) {
    float dg = deg[m0 + tid];
    sInv[tid] = 1.0f / fmaxf(dg, 1.0f);
    sNorm[tid] = 0.0f;
  }
  __syncthreads();

  // Cooperative tile load: mean = agg * (1/max(deg,1)), plus dst features.
  for (int i = tid; i < 16 * D; i += 256) {
    int r = i >> 7;
    int c = i & (D - 1);
    size_t g = (size_t)(m0 + r) * D + c;
    sMean[r][c] = agg[g] * sInv[r];
    sDst[r][c] = xdst[g];
  }
  __syncthreads();

  // WMMA fragment addressing (f32 16x16x4):
  //   A: lane m=lane&15 holds {A[m][4k+k0], A[m][4k+k0+1]}, k0 = (lane>=16)?2:0
  //   B: lane n=lane&15 holds {B[4k+k0][n], B[4k+k0+1][n]} = same pattern on W rows
  const int mrow = lane & 15;
  const int k0 = (lane >> 4) << 1;  // 0 or 2
  const int ncol = w * 16 + (lane & 15);

  // Accumulator initialized with bias (n is constant per lane across acc VGPRs).
  float bv = bl[ncol];
  v8f acc = {bv, bv, bv, bv, bv, bv, bv, bv};

  const float* WlRow = Wl + (size_t)ncol * D;
  const float* WrRow = Wr + (size_t)ncol * D;

#pragma unroll
  for (int kk = 0; kk < D / 4; ++kk) {
    v2f a = *(const v2f*)&sMean[mrow][4 * kk + k0];
    v2f b = *(const v2f*)&WlRow[4 * kk + k0];
    acc = __builtin_amdgcn_wmma_f32_16x16x4_f32(false, a, false, b,
                                                (short)0, acc, false, false);
  }
#pragma unroll
  for (int kk = 0; kk < D / 4; ++kk) {
    v2f a = *(const v2f*)&sDst[mrow][4 * kk + k0];
    v2f b = *(const v2f*)&WrRow[4 * kk + k0];
    acc = __builtin_amdgcn_wmma_f32_16x16x4_f32(false, a, false, b,
                                                (short)0, acc, false, false);
  }

  // C/D layout: VGPR v -> row (v + (lane>=16 ? 8 : 0)), col = lane&15 (in tile).
  const int rbase = (lane >> 4) << 3;
#pragma unroll
  for (int v = 0; v < 8; ++v) {
    sOut[rbase + v][ncol & (D - 1)] = acc[v];
  }
  __syncthreads();

  // L2 norm per row: 16 threads/row, 8 elements each, reduce via LDS atomics.
  {
    int r = tid >> 4;
    int cbase = (tid & 15) * 8;
    float p = 0.0f;
#pragma unroll
    for (int j = 0; j < 8; ++j) {
      float x = sOut[r][cbase + j];
      p += x * x;
    }
    atomicAdd(&sNorm[r], p);
  }
  __syncthreads();

  // Normalize (+relu) and write out, coalesced.
  for (int i = tid; i < 16 * D; i += 256) {
    int r = i >> 7;
    int c = i & (D - 1);
    float nrm = fmaxf(sqrtf(sNorm[r]), 1e-12f);
    float val = sOut[r][c] / nrm;
    if (RELU) val = fmaxf(val, 0.0f);
    out[(size_t)(m0 + r) * D + c] = val;
  }
}

// ---------------------------------------------------------------------------
// Launch: zero(agg/deg) -> scatter1 -> gemm1(relu->h) -> scatter2 -> gemm2(out)
// Workspace layout (floats): [agg1 | deg1 | agg2 | deg2 | h]
// ---------------------------------------------------------------------------
extern "C" void kernel_launch(void* const* d_in, const int* in_sizes, int n_in,
                              void* d_out, int out_size, void* d_ws,
                              size_t ws_size, hipStream_t stream) {
  const float* x    = (const float*)d_in[0];
  const float* W_l1 = (const float*)d_in[1];
  const float* b_l1 = (const float*)d_in[2];
  const float* W_r1 = (const float*)d_in[3];
  const float* W_l2 = (const float*)d_in[4];
  const float* b_l2 = (const float*)d_in[5];
  const float* W_r2 = (const float*)d_in[6];
  const int* row1 = (const int*)d_in[7];
  const int* col1 = (const int*)d_in[8];
  const int* row2 = (const int*)d_in[9];
  const int* col2 = (const int*)d_in[10];
  const int E1 = in_sizes[7];
  const int E2 = in_sizes[9];

  float* ws = (float*)d_ws;
  float* agg1 = ws;
  float* deg1 = agg1 + (size_t)N_MID * D;
  float* agg2 = deg1 + N_MID;
  float* deg2 = agg2 + (size_t)N_DST * D;
  float* h    = deg2 + N_DST;
  const size_t zero_floats =
      (size_t)N_MID * D + N_MID + (size_t)N_DST * D + N_DST;

  hipMemsetAsync(d_ws, 0, zero_floats * sizeof(float), stream);

  // Layer 1
  sage_scatter<<<(E1 + 7) / 8, 256, 0, stream>>>(x, row1, col1, agg1, deg1, E1);
  sage_gemm<1><<<N_MID / 16, 256, 0, stream>>>(agg1, deg1, x, W_l1, b_l1, W_r1,
                                               h, N_MID);
  // Layer 2
  sage_scatter<<<(E2 + 7) / 8, 256, 0, stream>>>(h, row2, col2, agg2, deg2, E2);
  sage_gemm<0><<<N_DST / 16, 256, 0, stream>>>(agg2, deg2, h, W_l2, b_l2, W_r2,
                                               (float*)d_out, N_DST);
}